// CELossWithSVLS_V7_26723286516276
// MI455X (gfx1250) — compile-verified
//
#include <hip/hip_runtime.h>
#include <hip/hip_bf16.h>
#include <stdint.h>

// Problem dims (fixed by reference setup_inputs)
#define NN 2
#define CC 8
#define ZZ 64
#define XX 128
#define YY 128
#define ZXY (ZZ*XX*YY)            // 1,048,576
#define NVOX (NN*ZXY)             // 2,097,152

// Tile: 2 x 16 x 64 voxels per block, 256 threads, 8 voxels/thread
#define TZ 2
#define TX 16
#define TY 64
#define HZ (TZ+2)
#define HX (TX+2)
#define HY (TY+2)
#define HXY (HX*HY)
#define HALO (HZ*HX*HY)           // 4*18*66 = 4752
#define NTHREADS 256
#define VPT ((TZ*TX*TY)/NTHREADS) // 8
#define NB ((ZZ/TZ)*(XX/TX)*(YY/TY)*NN) // 32*8*2*2 = 1024 blocks

typedef __attribute__((ext_vector_type(2))) float v2f;
typedef __attribute__((ext_vector_type(8))) float v8f;

#if defined(__has_builtin)
#if __has_builtin(__builtin_amdgcn_global_load_async_to_lds_b32)
#define USE_ASYNC 1
#endif
#endif
#ifndef USE_ASYNC
#define USE_ASYNC 0
#endif

typedef __attribute__((address_space(1))) int g_i32;
typedef __attribute__((address_space(3))) int l_i32;

// Single v_rcp_f32 instead of the ~10-op IEEE division expansion.
__device__ __forceinline__ float fast_rcp(float x) {
#if defined(__has_builtin)
#if __has_builtin(__builtin_amdgcn_rcpf)
  return __builtin_amdgcn_rcpf(x);
#else
  return 1.0f / x;
#endif
#else
  return 1.0f / x;
#endif
}

// Wave32 reduction via V_WMMA_F32_16X16X4_F32:
// A(16x4): lane m<16 holds A[m,0]=p, A[m,1]=0; lane m>=16 holds A[m-16,2]=p, A[m-16,3]=0.
// B(4x16) = ones -> D[m][n] = rowsum(A)_m. Lanes 0..15 hold D[0..7][n] in 8 VGPRs,
// lanes 16..31 hold D[8..15][n]. Sum own 8 regs + xor-16 shuffle = full wave sum.
__device__ __forceinline__ float wave_wmma_reduce(float p) {
  v2f a; a[0] = p;    a[1] = 0.0f;
  v2f b; b[0] = 1.0f; b[1] = 1.0f;
  v8f c = {};
  v8f d = __builtin_amdgcn_wmma_f32_16x16x4_f32(false, a, false, b,
                                                (short)0, c, false, false);
  float s8 = d[0]+d[1]+d[2]+d[3]+d[4]+d[5]+d[6]+d[7];
  return s8 + __shfl_xor(s8, 16, 32);
}

__global__ __launch_bounds__(NTHREADS)
void svls_ce_kernel(const float* __restrict__ inputs,
                    const int*   __restrict__ labels,
                    const float* __restrict__ images,
                    float* __restrict__ partials) {
  __shared__ float s_img[HALO];
  __shared__ int   s_lab[HALO];
  __shared__ float s_logit[NTHREADS * CC];
  __shared__ float s_red[NTHREADS / 32];

  const int tid = threadIdx.x;
  const int b   = blockIdx.x;
  const int tyi = b & 1;
  const int txi = (b >> 1) & 7;
  const int tzi = (b >> 4) & 31;
  const int n   = b >> 9;
  const int z0 = tzi * TZ, x0 = txi * TX, y0 = tyi * TY;

  const float* img = images + (size_t)(n * 2 + 1) * ZXY;  // channel 1
  const int*   lab = labels + (size_t)n * ZXY;
  const float* inp = inputs + (size_t)n * CC * ZXY;

  // ---- Stage halo slabs (edge-clamped) into LDS, async on CDNA5 ----
  for (int i = tid; i < HALO; i += NTHREADS) {
    int hz = i / HXY;
    int r  = i - hz * HXY;
    int hx = r / HY;
    int hy = r - hx * HY;
    int gz = min(max(z0 + hz - 1, 0), ZZ - 1);
    int gx = min(max(x0 + hx - 1, 0), XX - 1);
    int gy = min(max(y0 + hy - 1, 0), YY - 1);
    int gidx = (gz * XX + gx) * YY + gy;
#if USE_ASYNC
    __builtin_amdgcn_global_load_async_to_lds_b32(
        (g_i32*)(img + gidx), (l_i32*)(&s_img[i]), 0, 0);
    __builtin_amdgcn_global_load_async_to_lds_b32(
        (g_i32*)(lab + gidx), (l_i32*)(&s_lab[i]), 0, 0);
#else
    s_img[i] = img[gidx];
    s_lab[i] = lab[gidx];
#endif
  }
#if USE_ASYNC
#if __has_builtin(__builtin_amdgcn_s_wait_asynccnt)
  __builtin_amdgcn_s_wait_asynccnt(0);
#else
  asm volatile("s_wait_asynccnt 0x0" ::: "memory");
#endif
#endif
  __syncthreads();

  float acc = 0.0f;
  for (int v = 0; v < VPT; ++v) {
    int li = tid + v * NTHREADS;
    int lz = li / (TX * TY);
    int r  = li - lz * (TX * TY);
    int lx = r / TY;
    int ly = r - lx * TY;
    int sp = ((z0 + lz) * XX + (x0 + lx)) * YY + (y0 + ly);

    // 8 logits: streaming, coalesced along y; log-sum-exp
    float xl[CC];
    float m = -3.4e38f;
#pragma unroll
    for (int c = 0; c < CC; ++c) {
      xl[c] = inp[c * ZXY + sp];
      m = fmaxf(m, xl[c]);
    }
    float se = 0.0f;
#pragma unroll
    for (int c = 0; c < CC; ++c) {
      se += __expf(xl[c] - m);
      s_logit[(tid << 3) + c] = xl[c];   // own slot: no barrier needed
    }
    float lse = m + __logf(se);

    // 27-point stencil from LDS halo
    int   ch   = (lz + 1) * HXY + (lx + 1) * HY + (ly + 1);
    float cimg = s_img[ch];
    float S = 0.0f, A = 0.0f;
#pragma unroll
    for (int dz = -1; dz <= 1; ++dz)
#pragma unroll
      for (int dx = -1; dx <= 1; ++dx)
#pragma unroll
        for (int dy = -1; dy <= 1; ++dy) {
          int   h    = ch + dz * HXY + dx * HY + dy;
          float diff = s_img[h] - cimg;
          float d2   = (float)(dz * dz + dx * dx + dy * dy);
          float u    = __expf(-0.5f * (diff * diff + d2)); // center -> 1
          int   lbl  = s_lab[h];
          float xv   = s_logit[(tid << 3) + lbl];          // indexed LDS read
          S += u;
          A  = fmaf(u, xv, A);
        }
    int   lc   = s_lab[ch];
    float xc   = s_logit[(tid << 3) + lc];

    float invS = fast_rcp(S);                   // v_rcp_f32
    float ns   = 1.0f - invS + 1e-6f;           // replaced center weight
    float rns  = fast_rcp(ns);
    float inv  = invS * rns;                    // 1/(S*ns)
    float sum2 = 2.0f - 1e-6f * rns;            // final normalizer
    float val  = (xc + (A - xc) * inv) * fast_rcp(sum2);
    acc += lse - val;                           // sum_k Wfinal = 1
  }

  // ---- wave reduce (WMMA), then block reduce, one partial per block ----
  float wsum = wave_wmma_reduce(acc);
  int wid = tid >> 5;
  if ((tid & 31) == 0) s_red[wid] = wsum;
  __syncthreads();
  if (tid == 0) {
    float t = 0.0f;
#pragma unroll
    for (int i = 0; i < NTHREADS / 32; ++i) t += s_red[i];
    partials[blockIdx.x] = t;
  }
}

__global__ __launch_bounds__(NTHREADS)
void svls_finalize_kernel(const float* __restrict__ partials,
                          float* __restrict__ out) {
  __shared__ float s_red[NTHREADS / 32];
  const int tid = threadIdx.x;
  float p = 0.0f;
  for (int i = tid; i < NB; i += NTHREADS) p += partials[i];
  float wsum = wave_wmma_reduce(p);
  int wid = tid >> 5;
  if ((tid & 31) == 0) s_red[wid] = wsum;
  __syncthreads();
  if (tid == 0) {
    float t = 0.0f;
#pragma unroll
    for (int i = 0; i < NTHREADS / 32; ++i) t += s_red[i];
    out[0] = t * (1.0f / (float)NVOX);
  }
}

extern "C" void kernel_launch(void* const* d_in, const int* in_sizes, int n_in,
                              void* d_out, int out_size, void* d_ws, size_t ws_size,
                              hipStream_t stream) {
  (void)in_sizes; (void)n_in; (void)out_size; (void)ws_size;
  const float* inputs = (const float*)d_in[0];
  const int*   labels = (const int*)d_in[1];
  const float* images = (const float*)d_in[2];
  float* partials = (float*)d_ws;   // NB floats, fully rewritten every call

  svls_ce_kernel<<<NB, NTHREADS, 0, stream>>>(inputs, labels, images, partials);
  svls_finalize_kernel<<<1, NTHREADS, 0, stream>>>(partials, (float*)d_out);
}